// Metric_35588099015209
// MI455X (gfx1250) — compile-verified
//
#include <hip/hip_runtime.h>

#define IN_DIM 9801
#define OUT_DIM 25
#define BS 64

typedef __attribute__((ext_vector_type(2)))  float v2f;
typedef __attribute__((ext_vector_type(8)))  float v8f;

// ---------------------------------------------------------------------------
// Kernel 1: initialize d with the K-tail term (k = 9800), so the WMMA kernel
// only has to cover k in [0, 9800), which is an exact multiple of 4.
// (Harness poisons d_out, and the WMMA kernel accumulates with atomics, so
// this doubles as the required initialization.)
// ---------------------------------------------------------------------------
__global__ void init_d_tail_kernel(const float* __restrict__ x,
                                   const float* __restrict__ measure,
                                   float* __restrict__ d) {
    int i = blockIdx.x * blockDim.x + threadIdx.x;
    if (i >= BS * OUT_DIM) return;
    const int b = i / OUT_DIM;
    const int o = i % OUT_DIM;
    d[i] = x[b * IN_DIM + (IN_DIM - 1)] * measure[o * IN_DIM + (IN_DIM - 1)];
}

// ---------------------------------------------------------------------------
// Kernel 2: d(64x25) += x(64x9800-slab) . measure^T via V_WMMA_F32_16X16X4.
// 8 waves/WG = 4 M-tiles x 2 N-tiles; split-K over gridDim.x WGs with f32
// global atomics. Inner loop is branch-free: all loads unconditional (K range
// is in-bounds by construction; invalid B rows use a clamped pointer + cndmask).
// ---------------------------------------------------------------------------
__global__ void __launch_bounds__(256) d_wmma_kernel(const float* __restrict__ x,
                                                     const float* __restrict__ measure,
                                                     float* __restrict__ dout,
                                                     int steps_per_wg,
                                                     int total_steps) {
    const int lane = threadIdx.x & 31;
    const int wave = threadIdx.x >> 5;   // 0..7
    const int lm   = lane & 15;          // 0..15
    const int lh   = lane >> 4;          // 0 or 1
    const int m0   = (wave & 3) << 4;    // 0,16,32,48  (batch tile)
    const int n0   = (wave >> 2) << 4;   // 0 or 16     (output-dim tile)

    const int s0 = blockIdx.x * steps_per_wg;
    const int s1 = min(s0 + steps_per_wg, total_steps);

    const int  am     = m0 + lm;                 // x row, always < 64
    const int  bn     = n0 + lm;                 // measure row, may be >= 25
    const bool bvalid = (bn < OUT_DIM);
    // A/B VGPR layout for 16x16x4 f32: K = vgpr + 2*(lane/16)
    const float* ap = x + am * IN_DIM + 2 * lh;
    const float* bp = measure + (bvalid ? bn : (OUT_DIM - 1)) * IN_DIM + 2 * lh;

    v8f c = {};
#pragma unroll 4
    for (int s = s0; s < s1; ++s) {
        const int k = s * 4;                     // k in [0, 9800): always in-bounds
        v2f a, b;
        a[0] = ap[k];
        a[1] = ap[k + 1];
        const float b0 = bp[k];
        const float b1 = bp[k + 1];
        b[0] = bvalid ? b0 : 0.0f;               // v_cndmask, no branch
        b[1] = bvalid ? b1 : 0.0f;
        c = __builtin_amdgcn_wmma_f32_16x16x4_f32(false, a, false, b,
                                                  (short)0, c, false, false);
    }

#pragma unroll
    for (int v = 0; v < 8; ++v) {
        const int M = m0 + v + 8 * lh;           // C/D layout: M = vgpr + 8*(lane/16)
        const int N = n0 + lm;                   // N = lane%16
        if (N < OUT_DIM) atomicAdd(&dout[M * OUT_DIM + N], c[v]);
    }
}

// ---------------------------------------------------------------------------
// Kernel 3: d_full[b,f] = fullmeasure[f,f] * x[b,f].
// Reads only the diagonal of the 384MB matrix (~1.25MB of cache lines),
// streams x once (NT load) and the output once (NT store): ~6MB total
// => ~0.3us at 23.3 TB/s vs 16.5us for the naive full-matrix read.
// ---------------------------------------------------------------------------
__global__ void __launch_bounds__(256) dfull_mask_kernel(const float* __restrict__ x,
                                                         const float* __restrict__ fullmeasure,
                                                         float* __restrict__ out) {
    const int i = blockIdx.x * blockDim.x + threadIdx.x;   // < 64*9801 = 627264
    if (i >= BS * IN_DIM) return;
    const int f = i % IN_DIM;                              // column within row
    const float diag = fullmeasure[f * IN_DIM + f];        // max idx ~96M < 2^31
    const float xv   = __builtin_nontemporal_load(&x[i]);  // streamed once
    __builtin_nontemporal_store(diag * xv, &out[i]);       // streamed once
}

// ---------------------------------------------------------------------------
extern "C" void kernel_launch(void* const* d_in, const int* in_sizes, int n_in,
                              void* d_out, int out_size, void* d_ws, size_t ws_size,
                              hipStream_t stream) {
    const float* x           = (const float*)d_in[0];  // (64, 9801)
    const float* measure     = (const float*)d_in[1];  // (25, 9801)
    const float* fullmeasure = (const float*)d_in[2];  // (9801, 9801)

    float* dptr  = (float*)d_out;                // first 64*25 floats
    float* dfull = dptr + BS * OUT_DIM;          // next 64*9801 floats

    // 1) init d with the k=9800 tail term (also serves as the zero-init)
    init_d_tail_kernel<<<(BS * OUT_DIM + 255) / 256, 256, 0, stream>>>(
        x, measure, dptr);

    // 2) WMMA split-K GEMM over k in [0, 9800) (exact multiple of 4)
    const int SPLITK       = 64;
    const int total_steps  = (IN_DIM - 1) / 4;   // 2450 full K-steps
    const int steps_per_wg = (total_steps + SPLITK - 1) / SPLITK;  // 39
    d_wmma_kernel<<<SPLITK, 256, 0, stream>>>(x, measure, dptr,
                                              steps_per_wg, total_steps);

    // 3) diagonal-mask streaming kernel for d_full
    dfull_mask_kernel<<<(BS * IN_DIM + 255) / 256, 256, 0, stream>>>(
        x, fullmeasure, dfull);
}